// FocalLossAdaptive_6030134084185
// MI455X (gfx1250) — compile-verified
//
#include <hip/hip_runtime.h>
#include <hip/hip_bf16.h>

// Focal loss (adaptive gamma), N=32768 rows x C=4096 classes, f32.
// HBM-bound: 512 MiB read once -> ~22 us floor at 23.3 TB/s.
// CDNA5 paths: global_load_async_to_lds_b128 (row staging), s_wait_asynccnt,
// v_wmma_f32_16x16x4_f32 (cross-lane sum reduction).

#define TILE_C 4096
#define BLK    256
#define NWAVE  (BLK / 32)

typedef __attribute__((ext_vector_type(2))) float v2f;
typedef __attribute__((ext_vector_type(8))) float v8f;
typedef int v4i __attribute__((vector_size(16)));   // matches builtin param pointee

#if defined(__gfx1250__) && __has_builtin(__builtin_amdgcn_wmma_f32_16x16x4_f32)
#define HAVE_WMMA_F32_REDUCE 1
#endif

// Sum of v over all 32 lanes; result valid in every lane.
// Uses V_WMMA_F32_16X16X4_F32 with B = ones:
//   A layout (16x4 f32): lane L<16 -> A[L][0]=vgpr0, A[L][1]=vgpr1;
//                        lane L>=16 -> A[L-16][2]=vgpr0, A[L-16][3]=vgpr1.
//   With a=(v,0): D[m][n] = v(m) + v(m+16)  (column-uniform).
//   D layout: vgpr r holds row M=r (lanes 0-15) / M=8+r (lanes 16-31),
//   so sum(d[0..7]) = half-wave sum; shfl_xor(16) completes it.
__device__ __forceinline__ float wave_reduce_sum(float v) {
#if defined(HAVE_WMMA_F32_REDUCE)
  v2f a; a[0] = v;    a[1] = 0.0f;
  v2f b; b[0] = 1.0f; b[1] = 1.0f;
  v8f c = {};
  v8f d = __builtin_amdgcn_wmma_f32_16x16x4_f32(
      /*neg_a=*/false, a, /*neg_b=*/false, b,
      /*c_mod=*/(short)0, c, /*reuse_a=*/false, /*reuse_b=*/false);
  float s = ((d[0] + d[1]) + (d[2] + d[3])) + ((d[4] + d[5]) + (d[6] + d[7]));
  s += __shfl_xor(s, 16, 32);
  return s;
#else
  #pragma unroll
  for (int off = 16; off; off >>= 1) v += __shfl_xor(v, off, 32);
  return v;
#endif
}

__global__ void __launch_bounds__(BLK)
focal_row_kernel(const float* __restrict__ inp,
                 const int*   __restrict__ tgt,
                 float*       __restrict__ rowloss) {
  __shared__ float xr[TILE_C];
  __shared__ float red[NWAVE];

  const int tid  = threadIdx.x;
  const int lane = tid & 31;
  const int wid  = tid >> 5;
  const int row  = blockIdx.x;
  const float* rp = inp + (size_t)row * TILE_C;

  // ---- Stage the 16KB row into LDS via async DMA (ASYNCcnt-tracked) ----
#if defined(__gfx1250__) && __has_builtin(__builtin_amdgcn_global_load_async_to_lds_b128)
  {
    // Builtin signature (from hipcc diagnostic): non-const int4* in AS1, int4* in AS3.
    __attribute__((address_space(1))) v4i* g =
        (__attribute__((address_space(1))) v4i*)rp;
    __attribute__((address_space(3))) v4i* l =
        (__attribute__((address_space(3))) v4i*)xr;
    #pragma unroll
    for (int i = 0; i < (TILE_C * 4) / (BLK * 16); ++i) {  // 4 x 4KB chunks
      const int e = i * BLK + tid;                          // 16-byte elements
      __builtin_amdgcn_global_load_async_to_lds_b128(g + e, l + e, 0, 0);
    }
  }
#else
  #pragma unroll
  for (int i = 0; i < TILE_C / (BLK * 4); ++i) {
    const int idx = (i * BLK + tid) * 4;
    *(float4*)&xr[idx] = *(const float4*)&rp[idx];
  }
#endif
  asm volatile("s_wait_asynccnt 0x0" ::: "memory");  // own transfers done (NOP if none)
  __syncthreads();                                   // all waves' transfers visible

  // ---- Pull 16 elements/thread into registers (conflict-free stride) ----
  float x[TILE_C / BLK];
  #pragma unroll
  for (int i = 0; i < TILE_C / BLK; ++i) x[i] = xr[i * BLK + tid];

  // ---- Row max ----
  float m = x[0];
  #pragma unroll
  for (int i = 1; i < TILE_C / BLK; ++i) m = fmaxf(m, x[i]);
  #pragma unroll
  for (int off = 16; off; off >>= 1) m = fmaxf(m, __shfl_xor(m, off, 32));
  if (lane == 0) red[wid] = m;
  __syncthreads();
  float M = red[0];
  #pragma unroll
  for (int w = 1; w < NWAVE; ++w) M = fmaxf(M, red[w]);
  __syncthreads();  // red is about to be reused

  // ---- Row sum of exp(x - M) ----
  float s = 0.0f;
  #pragma unroll
  for (int i = 0; i < TILE_C / BLK; ++i) s += __expf(x[i] - M);
  s = wave_reduce_sum(s);            // WMMA-based, whole wave active (EXEC all-1s)
  if (lane == 0) red[wid] = s;
  __syncthreads();

  // ---- Per-row focal loss ----
  if (tid == 0) {
    float S = red[0];
    #pragma unroll
    for (int w = 1; w < NWAVE; ++w) S += red[w];
    const int   t     = tgt[row];
    const float logpt = xr[t] - M - __logf(S);
    const float pt    = __expf(logpt);
    const float u     = 1.0f - pt;
    float w;
    if (pt >= 0.5f)       w = 1.0f;                       // gamma = 0
    else if (pt < 0.2f) { const float u2 = u * u; w = u2 * u2 * u; }  // gamma = 5
    else                  w = u * u * u;                  // gamma = 3
    rowloss[row] = -w * logpt;
  }
}

__global__ void __launch_bounds__(BLK)
final_reduce_kernel(const float* __restrict__ part, int n, float* __restrict__ out) {
  __shared__ float red[NWAVE];
  const int tid = threadIdx.x;
  const int lane = tid & 31;
  const int wid  = tid >> 5;
  float s = 0.0f;
  for (int i = tid; i < n; i += BLK) s += part[i];
  s = wave_reduce_sum(s);
  if (lane == 0) red[wid] = s;
  __syncthreads();
  if (tid == 0) {
    float S = red[0];
    #pragma unroll
    for (int w = 1; w < NWAVE; ++w) S += red[w];
    out[0] = S;
  }
}

extern "C" void kernel_launch(void* const* d_in, const int* in_sizes, int n_in,
                              void* d_out, int out_size, void* d_ws, size_t ws_size,
                              hipStream_t stream) {
  const float* inp = (const float*)d_in[0];  // [N, 4096] f32
  const int*   tgt = (const int*)d_in[1];    // [N] i32
  const int N = in_sizes[1];

  float* rowloss = (float*)d_ws;             // N floats of scratch (128 KB)

  focal_row_kernel<<<N, BLK, 0, stream>>>(inp, tgt, rowloss);
  final_reduce_kernel<<<1, BLK, 0, stream>>>(rowloss, N, (float*)d_out);
}